// EVRP_Solver_64819646431493
// MI455X (gfx1250) — compile-verified
//
#include <hip/hip_runtime.h>

// ---------------- problem constants (fixed by the reference) ----------------
#define BS     256
#define SEQN   256
#define HN     128
#define H2     256   // 2H
#define H4     512   // 4H
#define STEPS  128
#define NUM_AFS 3
#define NEGV  (-1000000000.0f)

typedef __attribute__((ext_vector_type(2))) float v2f;
typedef __attribute__((ext_vector_type(8))) float v8f;

// ---------------- fast math (v_exp_f32 / v_log_f32 / v_rcp_f32) -------------
__device__ __forceinline__ float fexp(float x) {
  return __builtin_amdgcn_exp2f(x * 1.4426950408889634f);
}
__device__ __forceinline__ float flog(float x) {
  return __builtin_amdgcn_logf(x) * 0.6931471805599453f;
}
__device__ __forceinline__ float fsig(float x) {
  return __builtin_amdgcn_rcpf(1.0f + fexp(-x));
}
__device__ __forceinline__ float ftanh(float x) {
  float xx = fminf(fmaxf(x, -15.0f), 15.0f);
  float e  = __builtin_amdgcn_exp2f(xx * -2.8853900817779268f); // exp(-2x)
  return (1.0f - e) * __builtin_amdgcn_rcpf(1.0f + e);
}
__device__ __forceinline__ unsigned hash3(unsigned a, unsigned b, unsigned c) {
  unsigned h = (a * 0x9E3779B1u) ^ (b * 0x85EBCA77u) ^ (c * 0xC2B2AE3Du) ^ 0x27D4EB2Fu;
  h ^= h >> 15; h *= 0x2C1B3C6Du; h ^= h >> 12; h *= 0x297A2D39u; h ^= h >> 15;
  return h;
}

// ---------------- WMMA f32 16x16x4 wrapper ----------------------------------
// A: 16x4 (lane m = lane&15, K pair = 2*(lane>>4); vgpr j holds K=pair+j)
// B:  4x16 (lane n = lane&15, same K striping)
// C/D: vgpr r, lane l -> (M = r + 8*(l>>4), N = l&15)
__device__ __forceinline__ v8f wmma4(v2f a, v2f b, v8f c) {
  return __builtin_amdgcn_wmma_f32_16x16x4_f32(false, a, false, b, (short)0, c,
                                               false, false);
}

// ============================================================================
// Init: demands, mask, dis_by_afs, fuel/time, old_idx, cst=0, xcat=0, din2=0
// grid = BS*SEQN/256
__global__ void k_init(const float* __restrict__ dynamic,
                       const float* __restrict__ dist,
                       float* dem, float* maskf, float* dis_afs,
                       float* fuel, float* timev, int* old_idx,
                       float* cst, float* xcat, float* din2) {
  int i = blockIdx.x * blockDim.x + threadIdx.x;     // over BS*SEQN
  int s = i % SEQN, b = i / SEQN;
  dem[i]   = dynamic[(b * 3 + 2) * SEQN + s];
  maskf[i] = 1.0f;
  float m = 1e30f;
  #pragma unroll
  for (int a = 1; a <= NUM_AFS; ++a) {
    float d = dist[(b * SEQN + a) * SEQN + 0] + dist[(b * SEQN + a) * SEQN + s];
    m = fminf(m, d);
  }
  dis_afs[i] = (s == 0) ? 0.0f : m;
  if (s == 0) {
    fuel[b]    = dynamic[(b * 3 + 0) * SEQN + 0];
    timev[b]   = dynamic[(b * 3 + 1) * SEQN + 0];
    old_idx[b] = 0;
  }
  if (i < BS * HN) cst[i] = 0.0f;
  if (i < BS * H2) { xcat[i] = 0.0f; din2[i] = 0.0f; }
}

// stat_emb[b,h,s] = W_s[h,:] . static[b,:,s] + b_s[h]; dec_in(0) = stat_emb[:,:,0]
// grid = BS*HN*SEQN/256
__global__ void k_stat_emb(const float* __restrict__ Ws,
                           const float* __restrict__ bs_,
                           const float* __restrict__ stat,
                           float* stat_emb, float* din2) {
  int i = blockIdx.x * blockDim.x + threadIdx.x;     // == (b*HN+h)*SEQN + s
  int s = i % SEQN, r = i / SEQN, h = r % HN, b = r / HN;
  float x0 = stat[(b * 2 + 0) * SEQN + s];
  float x1 = stat[(b * 2 + 1) * SEQN + s];
  float v  = Ws[h * 2 + 0] * x0 + Ws[h * 2 + 1] * x1 + bs_[h];
  stat_emb[i] = v;
  if (s == 0) din2[b * H2 + h] = v;                  // initial dec_in
}

// A3 = W_att[:,H:2H] @ W_d  (H x 3),  abias = W_att[:,H:2H] @ b_d
// grid = 1 block, 128 threads
__global__ void k_a3(const float* __restrict__ W_att,
                     const float* __restrict__ W_d,
                     const float* __restrict__ b_d,
                     float* A3, float* abias) {
  int k = threadIdx.x;
  float a0 = 0.f, a1 = 0.f, a2 = 0.f, ab = 0.f;
  for (int j = 0; j < HN; ++j) {
    float w = W_att[k * (3 * HN) + HN + j];
    a0 += w * W_d[j * 3 + 0];
    a1 += w * W_d[j * 3 + 1];
    a2 += w * W_d[j * 3 + 2];
    ab += w * b_d[j];
  }
  A3[k * 3 + 0] = a0; A3[k * 3 + 1] = a1; A3[k * 3 + 2] = a2; abias[k] = ab;
}

// pre_stat[b] = W_att[:,0:H] @ stat_emb[b]  (batched 128x256, K=128)  [WMMA]
// waves = BS*8*16; 128 thr/block -> 8192 blocks
__global__ void k_prestat(const float* __restrict__ W_att,
                          const float* __restrict__ stat_emb,
                          float* __restrict__ pre) {
  int wave = (blockIdx.x * blockDim.x + threadIdx.x) >> 5;
  int lane = threadIdx.x & 31;
  int tn = wave & 15, tm = (wave >> 4) & 7, b = wave >> 7;
  int ml = lane & 15, hi = lane >> 4;
  const float* Ar = W_att + (tm * 16 + ml) * (3 * HN);       // cols 0..H-1
  const float* Bc = stat_emb + (b * HN) * SEQN + (tn * 16 + ml);
  v8f acc = {0.f, 0.f, 0.f, 0.f, 0.f, 0.f, 0.f, 0.f};
  for (int k0 = 0; k0 < HN; k0 += 4) {
    int ka = k0 + 2 * hi;
    v2f a = {Ar[ka], Ar[ka + 1]};
    v2f bb = {Bc[ka * SEQN], Bc[(ka + 1) * SEQN]};
    acc = wmma4(a, bb, acc);
  }
  int m0 = tm * 16 + 8 * hi, n = tn * 16 + ml;
  float* Cp = pre + (b * HN + m0) * SEQN + n;
  #pragma unroll
  for (int r = 0; r < 8; ++r) Cp[r * SEQN] = acc[r];
}

// gates[n, 0:4H] = [dec_in|hst][n, 0:2H] @ [W_i|W_h]^T + b_i + b_h   [WMMA]
// M=512 N=256 K=256 -> 512 tiles -> 128 blocks of 128 threads
__global__ void k_gates(const float* __restrict__ W_i,
                        const float* __restrict__ W_h,
                        const float* __restrict__ b_i,
                        const float* __restrict__ b_h,
                        const float* __restrict__ din2,
                        float* __restrict__ gates) {
  int wave = (blockIdx.x * blockDim.x + threadIdx.x) >> 5;
  int lane = threadIdx.x & 31;
  int tn = wave & 15, tm = wave >> 4;
  int ml = lane & 15, hi = lane >> 4;
  int am = tm * 16 + ml;
  int bn = tn * 16 + ml;
  v8f acc = {0.f, 0.f, 0.f, 0.f, 0.f, 0.f, 0.f, 0.f};
  for (int k0 = 0; k0 < H2; k0 += 4) {
    int ka = k0 + 2 * hi;
    v2f a = {(ka < HN) ? W_i[am * HN + ka] : W_h[am * HN + ka - HN],
             (ka + 1 < HN) ? W_i[am * HN + ka + 1] : W_h[am * HN + ka + 1 - HN]};
    v2f bb = {din2[bn * H2 + ka], din2[bn * H2 + ka + 1]};
    acc = wmma4(a, bb, acc);
  }
  int m0 = tm * 16 + 8 * hi, n = tn * 16 + ml;
  #pragma unroll
  for (int r = 0; r < 8; ++r) {
    int m = m0 + r;
    gates[n * H4 + m] = acc[r] + b_i[m] + b_h[m];
  }
}

// LSTM pointwise: cst, h_lstm (-> xcat[:,H:2H]).  grid = BS*HN/256
__global__ void k_lstm(const float* __restrict__ gates,
                       float* cst, float* xcat) {
  int i = blockIdx.x * blockDim.x + threadIdx.x;
  int m = i % HN, b = i / HN;
  const float* g = gates + b * H4;
  float ig = fsig(g[m]);
  float fg = fsig(g[HN + m]);
  float gg = ftanh(g[2 * HN + m]);
  float og = fsig(g[3 * HN + m]);
  float c  = fg * cst[i] + ig * gg;
  cst[i] = c;
  xcat[b * H2 + HN + m] = og * ftanh(c);   // h_lstm
}

// hatt[n,m] = W_att[:,2H:3H] @ h_lstm  (M=128 N=256 K=128)  [WMMA]
__global__ void k_hatt(const float* __restrict__ W_att,
                       const float* __restrict__ xcat,
                       float* __restrict__ hatt) {
  int wave = (blockIdx.x * blockDim.x + threadIdx.x) >> 5;
  int lane = threadIdx.x & 31;
  int tn = wave & 15, tm = wave >> 4;
  int ml = lane & 15, hi = lane >> 4;
  const float* Ar = W_att + (tm * 16 + ml) * (3 * HN) + 2 * HN;
  const float* Bc = xcat + (tn * 16 + ml) * H2 + HN;
  v8f acc = {0.f, 0.f, 0.f, 0.f, 0.f, 0.f, 0.f, 0.f};
  for (int k0 = 0; k0 < HN; k0 += 4) {
    int ka = k0 + 2 * hi;
    v2f a  = {Ar[ka], Ar[ka + 1]};
    v2f bb = {Bc[ka], Bc[ka + 1]};
    acc = wmma4(a, bb, acc);
  }
  int m0 = tm * 16 + 8 * hi, n = tn * 16 + ml;
  #pragma unroll
  for (int r = 0; r < 8; ++r) hatt[n * HN + m0 + r] = acc[r];
}

// h_new = tanh(W_c @ [context|h_lstm] + b_c) -> din2[:,H:2H] (next hst)  [WMMA]
// M=128 N=256 K=256
__global__ void k_hnew(const float* __restrict__ W_c,
                       const float* __restrict__ b_c,
                       const float* __restrict__ xcat,
                       float* __restrict__ din2) {
  int wave = (blockIdx.x * blockDim.x + threadIdx.x) >> 5;
  int lane = threadIdx.x & 31;
  int tn = wave & 15, tm = wave >> 4;
  int ml = lane & 15, hi = lane >> 4;
  const float* Ar = W_c + (tm * 16 + ml) * H2;
  const float* Bc = xcat + (tn * 16 + ml) * H2;
  v8f acc = {0.f, 0.f, 0.f, 0.f, 0.f, 0.f, 0.f, 0.f};
  for (int k0 = 0; k0 < H2; k0 += 4) {
    int ka = k0 + 2 * hi;
    v2f a  = {Ar[ka], Ar[ka + 1]};
    v2f bb = {Bc[ka], Bc[ka + 1]};
    acc = wmma4(a, bb, acc);
  }
  int m0 = tm * 16 + 8 * hi, n = tn * 16 + ml;
  #pragma unroll
  for (int r = 0; r < 8; ++r) {
    int m = m0 + r;
    din2[n * H2 + HN + m] = ftanh(acc[r] + b_c[m]);
  }
}

// ============================================================================
// Fused attention / softmax / sample / context / state update.
// One block per batch row; 256 threads (one per node s).
__global__ void __launch_bounds__(256)
k_attn(const float* __restrict__ pre, const float* __restrict__ stat_emb,
       const float* __restrict__ dist, const float* __restrict__ hatt,
       const float* __restrict__ A3, const float* __restrict__ abias,
       const float* __restrict__ v_att, const float* __restrict__ dis_afs,
       float* dem, float* maskf, float* fuel, float* timev, int* old_idx,
       float* din2, float* xcat,
       float* out_probs, float* out_tours, float* out_logps, int step) {
  int b = blockIdx.x, t = threadIdx.x;
  __shared__ float bias_s[HN], v_s[HN], a2_s[HN];
  __shared__ float red[256];
  __shared__ int   redi[256];
  __shared__ float probs_s[SEQN];
  __shared__ float sh_fuel, sh_time, sh_chosen;
  __shared__ int   sh_idx;

  if (t < HN) {
    bias_s[t] = hatt[b * HN + t] + A3[t * 3 + 0] * fuel[b] +
                A3[t * 3 + 1] * timev[b] + abias[t];
    v_s[t]  = v_att[t];
    a2_s[t] = A3[t * 3 + 2];
  }
  __syncthreads();

  int s = t;
  float dm = dem[b * SEQN + s];
  const float* pp = pre + (b * HN) * SEQN + s;
  float e = 0.0f;
  #pragma unroll 4
  for (int k = 0; k < HN; ++k)
    e += v_s[k] * ftanh(pp[k * SEQN] + a2_s[k] * dm + bias_s[k]);

  float logit = (maskf[b * SEQN + s] > 0.5f) ? e : NEGV;

  // --- softmax (max, sum) ---
  red[t] = logit; __syncthreads();
  for (int off = 128; off > 0; off >>= 1) {
    if (t < off) red[t] = fmaxf(red[t], red[t + off]);
    __syncthreads();
  }
  float maxv = red[0]; __syncthreads();
  float ex = fexp(logit - maxv);
  red[t] = ex; __syncthreads();
  for (int off = 128; off > 0; off >>= 1) {
    if (t < off) red[t] += red[t + off];
    __syncthreads();
  }
  float sum = red[0]; __syncthreads();
  float p = ex * __builtin_amdgcn_rcpf(sum);
  probs_s[s] = p;
  out_probs[(size_t)(b * STEPS + step) * SEQN + s] = p;

  // --- Gumbel-max categorical sample (deterministic hash RNG) ---
  unsigned hsh = hash3((unsigned)step, (unsigned)b, (unsigned)s);
  float u = (float)((hsh >> 8) & 0xFFFFFFu) * (1.0f / 16777216.0f) + 1.0e-7f;
  float g = -flog(-flog(u));
  red[t] = logit + g; redi[t] = t; __syncthreads();
  for (int off = 128; off > 0; off >>= 1) {
    if (t < off && red[t + off] > red[t]) { red[t] = red[t + off]; redi[t] = redi[t + off]; }
    __syncthreads();
  }
  if (t == 0) sh_idx = redi[0];
  __syncthreads();
  int idx = sh_idx;
  if (t == idx) sh_chosen = logit;
  __syncthreads();

  // --- context = probs @ stat_emb^T ; dec_in = stat_emb[:, idx] ---
  if (t < HN) {
    const float* se = stat_emb + (size_t)(b * HN + t) * SEQN;
    float c = 0.0f;
    for (int ss = 0; ss < SEQN; ++ss) c += probs_s[ss] * se[ss];
    xcat[b * H2 + t] = c;
    din2[b * H2 + t] = se[idx];
  }

  // --- scalar state update ---
  if (t == 0) {
    int oi = old_idx[b];
    float dis = dist[(b * SEQN + oi) * SEQN + idx];
    float f  = fuel[b] - 0.2f * dis;
    float tm = timev[b] - dis * (1.0f / 60.0f);
    bool at_dep = (idx == 0);
    bool at_af  = (idx >= 1) && (idx <= NUM_AFS);
    if (at_dep || at_af) f = 60.0f;
    if (at_dep) tm = 11.0f;
    fuel[b] = f; timev[b] = tm; old_idx[b] = idx;
    sh_fuel = f; sh_time = tm;
    out_tours[b * STEPS + step] = (float)idx;
    out_logps[b * STEPS + step] = sh_chosen - (maxv + flog(sum));
  }
  __syncthreads();

  // --- demand + mask update (per node) ---
  float f = sh_fuel, tmv = sh_time;
  float dnew = dm;
  if (s == idx && idx > NUM_AFS) dnew = 0.0f;
  if (s == idx) dem[b * SEQN + s] = dnew;
  float d_next = dist[(b * SEQN + idx) * SEQN + s];
  bool fuel_ok = f >= 0.2f * (d_next + dis_afs[b * SEQN + s]);
  bool time_ok = tmv >= (d_next + dist[(b * SEQN + s) * SEQN + 0]) * (1.0f / 60.0f);
  bool cust = (dnew > 0.0f) && fuel_ok && time_ok;
  bool mk = (s <= NUM_AFS) ? (s != idx) : cust;
  if (s == 0) mk = true;
  maskf[b * SEQN + s] = mk ? 1.0f : 0.0f;
}

// ============================================================================
extern "C" void kernel_launch(void* const* d_in, const int* in_sizes, int n_in,
                              void* d_out, int out_size, void* d_ws, size_t ws_size,
                              hipStream_t stream) {
  (void)in_sizes; (void)n_in; (void)out_size; (void)ws_size;
  const float* stat    = (const float*)d_in[0];
  const float* dynamic = (const float*)d_in[1];
  const float* dist    = (const float*)d_in[2];
  const float* W_s     = (const float*)d_in[3];
  const float* b_s     = (const float*)d_in[4];
  const float* W_d     = (const float*)d_in[5];
  const float* b_d     = (const float*)d_in[6];
  const float* W_i     = (const float*)d_in[7];
  const float* W_h     = (const float*)d_in[8];
  const float* b_i     = (const float*)d_in[9];
  const float* b_h     = (const float*)d_in[10];
  const float* W_att   = (const float*)d_in[11];
  const float* v_att   = (const float*)d_in[12];
  const float* W_c     = (const float*)d_in[13];
  const float* b_c     = (const float*)d_in[14];
  // d_in[15] = decode_steps == 128 (compile-time STEPS)

  float* ws = (float*)d_ws;
  float* stat_emb = ws;                  // 8,388,608
  float* pre      = ws + 8388608;        // 8,388,608
  float* dis_afs  = ws + 16777216;       // 65,536
  float* dem      = ws + 16842752;       // 65,536
  float* maskf    = ws + 16908288;       // 65,536
  float* gates    = ws + 16973824;       // 131,072
  float* cst      = ws + 17104896;       // 32,768
  float* hatt     = ws + 17137664;       // 32,768
  float* din2     = ws + 17170432;       // 65,536  [dec_in | hst]
  float* xcat     = ws + 17235968;       // 65,536  [context | h_lstm]
  float* fuel     = ws + 17301504;       // 256
  float* timev    = ws + 17301760;       // 256
  float* A3       = ws + 17302016;       // 384
  float* abias    = ws + 17302400;       // 128
  int*   old_idx  = (int*)(ws + 17302528); // 256

  float* out_probs = (float*)d_out;                          // [BS,STEPS,SEQ]
  float* out_tours = out_probs + (size_t)BS * STEPS * SEQN;  // [BS,STEPS]
  float* out_logps = out_tours + (size_t)BS * STEPS;         // [BS,STEPS]

  k_init<<<BS * SEQN / 256, 256, 0, stream>>>(dynamic, dist, dem, maskf, dis_afs,
                                              fuel, timev, old_idx, cst, xcat, din2);
  k_stat_emb<<<BS * HN * SEQN / 256, 256, 0, stream>>>(W_s, b_s, stat, stat_emb, din2);
  k_a3<<<1, 128, 0, stream>>>(W_att, W_d, b_d, A3, abias);
  k_prestat<<<(BS * 8 * 16) / 4, 128, 0, stream>>>(W_att, stat_emb, pre);

  for (int step = 0; step < STEPS; ++step) {
    k_gates<<<(32 * 16) / 4, 128, 0, stream>>>(W_i, W_h, b_i, b_h, din2, gates);
    k_lstm<<<BS * HN / 256, 256, 0, stream>>>(gates, cst, xcat);
    k_hatt<<<(8 * 16) / 4, 128, 0, stream>>>(W_att, xcat, hatt);
    k_attn<<<BS, 256, 0, stream>>>(pre, stat_emb, dist, hatt, A3, abias, v_att,
                                   dis_afs, dem, maskf, fuel, timev, old_idx,
                                   din2, xcat, out_probs, out_tours, out_logps, step);
    k_hnew<<<(8 * 16) / 4, 128, 0, stream>>>(W_c, b_c, xcat, din2);
  }
}